// EmotionCaps_69320772157875
// MI455X (gfx1250) — compile-verified
//
#include <hip/hip_runtime.h>
#include <math.h>

typedef __attribute__((ext_vector_type(2))) float v2f;
typedef __attribute__((ext_vector_type(8))) float v8f;

#define B_TOT 256
#define C_TOT 1152
#define I_DIM 8
#define M_DIM 8
#define O_DIM 16
#define MO    128     // M*O
#define BQ    16      // batches per block (WMMA N dim)
#define CCH   64      // capsules per block
#define THREADS 256   // 8 wave32 waves

__device__ __forceinline__ v8f wmma_f32_16x16x4(v2f a, v2f b, v8f c) {
  // D = A(16x4 f32) * B(4x16 f32) + C(16x16 f32)
  return __builtin_amdgcn_wmma_f32_16x16x4_f32(
      /*neg_a=*/false, a, /*neg_b=*/false, b,
      /*c_mod=*/(short)0, c, /*reuse_a=*/false, /*reuse_b=*/false);
}

__device__ __forceinline__ float fast_rcp(float x) {
  return __builtin_amdgcn_rcpf(x);   // v_rcp_f32 (1 TRANS op vs ~12-op IEEE div)
}

// s-pass: s[b,m,o] += softmax_m(blog[b,c,:])[m] * u_hat[b,c,m,o], summed over this
// block's capsule chunk. u_hat recomputed via f32 WMMA (never materialized).
__global__ void __launch_bounds__(THREADS)
caps_spass(const float* __restrict__ u, const float* __restrict__ W,
           const float* __restrict__ blog, float* __restrict__ s_glob)
{
  __shared__ float uS[BQ * CCH * I_DIM];  // 32 KB: u[b_local][c_local][i]
  __shared__ float sS[MO * BQ];           //  8 KB: partial s[mo][b_local]
  const int tid = threadIdx.x;
  const int b0 = blockIdx.x * BQ;
  const int c0 = blockIdx.y * CCH;

  for (int k = tid; k < MO * BQ; k += THREADS) sS[k] = 0.0f;
  for (int k = tid; k < BQ * CCH * I_DIM; k += THREADS) {
    int bl = k >> 9;           // /(CCH*I_DIM)
    int rem = k & 511;
    uS[k] = u[(size_t)(b0 + bl) * (C_TOT * I_DIM) + (size_t)c0 * I_DIM + rem];
  }
  __syncthreads();

  const int wave = tid >> 5;   // wave32
  const int lane = tid & 31;
  const int col = lane & 15;   // N column = local batch index
  const int hi  = lane >> 4;   // half-wave: K pairs {0,1} vs {2,3}; rows 0-7 vs 8-15

  v8f zero = {};
  v8f acc[M_DIM];
  #pragma unroll
  for (int t = 0; t < M_DIM; ++t) acc[t] = zero;

  for (int cc = 0; cc < CCH / 8; ++cc) {
    const int c_local = wave * (CCH / 8) + cc;
    const int c = c0 + c_local;

    // B operand (4x16): U_c[i, b_local] from LDS, per-lane float2
    const float* up = &uS[(col * CCH + c_local) * I_DIM];
    const v2f bk0 = *(const v2f*)(up + 2 * hi);       // k = 0..3
    const v2f bk1 = *(const v2f*)(up + 4 + 2 * hi);   // k = 4..7

    // routing coefficients: softmax over m of blog[b0+col, c, :]
    const float4* bp = (const float4*)&blog[((size_t)(b0 + col) * C_TOT + c) * M_DIM];
    float4 l0 = bp[0], l1 = bp[1];
    float lg[8] = {l0.x, l0.y, l0.z, l0.w, l1.x, l1.y, l1.z, l1.w};
    float mx = lg[0];
    #pragma unroll
    for (int m = 1; m < 8; ++m) mx = fmaxf(mx, lg[m]);
    float ex[8]; float sum = 0.0f;
    #pragma unroll
    for (int m = 0; m < 8; ++m) { ex[m] = __expf(lg[m] - mx); sum += ex[m]; }
    const float inv = fast_rcp(sum);

    const float* Wc = &W[(size_t)c * (M_DIM * O_DIM * I_DIM)];
    #pragma unroll
    for (int t = 0; t < M_DIM; ++t) {  // tile t == m
      const float* wp = &Wc[t * (O_DIM * I_DIM) + col * I_DIM];
      const v2f a0 = *(const v2f*)(wp + 2 * hi);
      const v2f a1 = *(const v2f*)(wp + 4 + 2 * hi);
      v8f d = zero;
      d = wmma_f32_16x16x4(a0, bk0, d);
      d = wmma_f32_16x16x4(a1, bk1, d);
      const float cf = ex[t] * inv;        // coef[b=col, c, m=t]
      acc[t] += d * cf;
    }
  }

  // deposit per-wave accumulators into block-local s, then to global
  #pragma unroll
  for (int t = 0; t < M_DIM; ++t) {
    #pragma unroll
    for (int r = 0; r < 8; ++r) {
      int mo = t * 16 + hi * 8 + r;        // o = row index within tile
      atomicAdd(&sS[mo * BQ + col], acc[t][r]);
    }
  }
  __syncthreads();
  for (int k = tid; k < MO * BQ; k += THREADS) {
    int mo = k >> 4;
    int bl = k & 15;
    atomicAdd(&s_glob[(size_t)(b0 + bl) * MO + mo], sS[k]);
  }
}

// uv-pass: blog[b,c,m] += sum_o u_hat[b,c,m,o] * v[b,m,o] (u_hat via WMMA again)
__global__ void __launch_bounds__(THREADS)
caps_uvpass(const float* __restrict__ u, const float* __restrict__ W,
            const float* __restrict__ vin, float* __restrict__ blog)
{
  __shared__ float uS[BQ * CCH * I_DIM];  // 32 KB
  __shared__ float vS[BQ * MO];           //  8 KB: v[b_local][m*16+o]
  const int tid = threadIdx.x;
  const int b0 = blockIdx.x * BQ;
  const int c0 = blockIdx.y * CCH;

  for (int k = tid; k < BQ * CCH * I_DIM; k += THREADS) {
    int bl = k >> 9;
    int rem = k & 511;
    uS[k] = u[(size_t)(b0 + bl) * (C_TOT * I_DIM) + (size_t)c0 * I_DIM + rem];
  }
  for (int k = tid; k < BQ * MO; k += THREADS) {
    int bl = k >> 7;
    int mo = k & 127;
    vS[k] = vin[(size_t)(b0 + bl) * MO + mo];
  }
  __syncthreads();

  const int wave = tid >> 5;
  const int lane = tid & 31;
  const int col = lane & 15;
  const int hi  = lane >> 4;
  v8f zero = {};

  for (int cc = 0; cc < CCH / 8; ++cc) {
    const int c_local = wave * (CCH / 8) + cc;
    const int c = c0 + c_local;
    const float* up = &uS[(col * CCH + c_local) * I_DIM];
    const v2f bk0 = *(const v2f*)(up + 2 * hi);
    const v2f bk1 = *(const v2f*)(up + 4 + 2 * hi);
    const float* Wc = &W[(size_t)c * (M_DIM * O_DIM * I_DIM)];

    float uv[M_DIM];
    #pragma unroll
    for (int t = 0; t < M_DIM; ++t) {
      const float* wp = &Wc[t * (O_DIM * I_DIM) + col * I_DIM];
      const v2f a0 = *(const v2f*)(wp + 2 * hi);
      const v2f a1 = *(const v2f*)(wp + 4 + 2 * hi);
      v8f d = zero;
      d = wmma_f32_16x16x4(a0, bk0, d);
      d = wmma_f32_16x16x4(a1, bk1, d);
      float p = 0.0f;
      #pragma unroll
      for (int r = 0; r < 8; ++r)
        p += d[r] * vS[col * MO + t * 16 + hi * 8 + r];
      uv[t] = p;
    }
    // fold rows 0-7 (lanes<16) with rows 8-15 (lanes>=16); keep WMMA out of divergence
    #pragma unroll
    for (int t = 0; t < M_DIM; ++t) uv[t] += __shfl_xor(uv[t], 16, 32);
    if (lane < 16) {
      float* bp = &blog[((size_t)(b0 + col) * C_TOT + c) * M_DIM];
      #pragma unroll
      for (int t = 0; t < M_DIM; ++t) bp[t] += uv[t];
    }
  }
}

__global__ void caps_squash(const float* __restrict__ s, float* __restrict__ out)
{
  int g = blockIdx.x * blockDim.x + threadIdx.x;   // (b,m) pair
  if (g >= B_TOT * M_DIM) return;
  const float* sp = &s[(size_t)g * O_DIM];
  float n2 = 0.0f;
  #pragma unroll
  for (int o = 0; o < O_DIM; ++o) { float x = sp[o]; n2 += x * x; }
  float n = sqrtf(n2);
  float scale = n2 * __builtin_amdgcn_rcpf((1.0f + n2) * (n + 1e-8f));
  float* op = &out[(size_t)g * O_DIM];
  #pragma unroll
  for (int o = 0; o < O_DIM; ++o) op[o] = scale * sp[o];
}

__global__ void caps_zero(float* __restrict__ p, int n)
{
  int i = blockIdx.x * blockDim.x + threadIdx.x;
  if (i < n) p[i] = 0.0f;
}

extern "C" void kernel_launch(void* const* d_in, const int* in_sizes, int n_in,
                              void* d_out, int out_size, void* d_ws, size_t ws_size,
                              hipStream_t stream)
{
  const float* u = (const float*)d_in[0];   // [256,1152,8]
  const float* W = (const float*)d_in[1];   // [1152,8,16,8]
  float* blog = (float*)d_ws;                                 // [B,C,M] logits
  float* s    = blog + (size_t)B_TOT * C_TOT * M_DIM;          // [B,M,O]
  float* vbuf = s + (size_t)B_TOT * MO;                        // [B,M,O]
  float* outp = (float*)d_out;
  // ws bytes needed: (2359296 + 32768 + 32768)*4 ~= 9.3 MB

  const int NB = B_TOT * C_TOT * M_DIM;
  caps_zero<<<(NB + 255) / 256, 256, 0, stream>>>(blog, NB);

  dim3 grid(B_TOT / BQ, C_TOT / CCH);   // 16 x 18
  for (int r = 0; r < 3; ++r) {
    caps_zero<<<(B_TOT * MO + 255) / 256, 256, 0, stream>>>(s, B_TOT * MO);
    caps_spass<<<grid, THREADS, 0, stream>>>(u, W, blog, s);
    caps_squash<<<(B_TOT * M_DIM + 255) / 256, 256, 0, stream>>>(
        s, (r == 2) ? outp : vbuf);
    if (r < 2) caps_uvpass<<<grid, THREADS, 0, stream>>>(u, W, vbuf, blog);
  }
}